// Sample_11802570130409
// MI455X (gfx1250) — compile-verified
//
#include <hip/hip_runtime.h>
#include <stdint.h>

// Problem constants (from reference setup_inputs)
#define B_   16
#define N_   131072
#define P_   2048
#define WGS  1024
#define WGPB 8                          // workgroups cooperating per batch
#define PTS_PER_WG (N_ / WGPB)          // 16384 points per WG
#define PTS_PER_TH (PTS_PER_WG / WGS)   // 16 points per thread (register resident)

typedef unsigned long long u64;
typedef unsigned int u32;

__device__ __forceinline__ u64 kmax_u64(u64 a, u64 b) { return a > b ? a : b; }

__device__ __forceinline__ u64 shfl_xor_u64(u64 v, int lanemask) {
    u32 lo = (u32)v, hi = (u32)(v >> 32);
    lo = __shfl_xor(lo, lanemask, 32);
    hi = __shfl_xor(hi, lanemask, 32);
    return ((u64)hi << 32) | (u64)lo;
}

// ---------------------------------------------------------------------------
// Workspace init: best-key slots (atomicMax targets) and arrive counters must
// be zero at the start of every launch (graph replays included).
// ---------------------------------------------------------------------------
__global__ void fps_init(u64* __restrict__ best, u32* __restrict__ arrive) {
    int i = blockIdx.x * blockDim.x + threadIdx.x;
    if (i < B_ * P_) best[i] = 0ull;
    if (i < B_ * 16) arrive[i] = 0u;
}

// ---------------------------------------------------------------------------
// FPS: 8 WGs per batch, xyz + dist fully register resident.
// Per iteration: VALU distance update + argmax key, wave reduce, LDS reduce,
// one global atomicMax per WG, one inter-WG sync per batch (monotonic counter).
// ---------------------------------------------------------------------------
__global__ __launch_bounds__(WGS) void fps_kernel(const float* __restrict__ pts,
                                                  u64* __restrict__ best,
                                                  u32* __restrict__ arrive) {
    __shared__ float slab[3 * PTS_PER_WG];   // 192 KB staging for xyz
    __shared__ u64   wred[WGS / 32];         // per-wave argmax partials
    __shared__ float scent[3];               // broadcast centroid

    const int b    = blockIdx.x / WGPB;
    const int w    = blockIdx.x % WGPB;
    const int tid  = threadIdx.x;
    const int baseN = w * PTS_PER_WG;
    const float* bpts = pts + (size_t)b * 3 * N_;

    // ---- async bulk copy: global -> LDS (gfx1250 ASYNCcnt path) ----
    // Each channel slice is PTS_PER_WG floats = 64 KB = 4096 x 16B chunks;
    // 1024 threads issue 4 chunks each, fully coalesced.
    {
        unsigned slab_lds = (unsigned)(uintptr_t)(&slab[0]);
#pragma unroll
        for (int c = 0; c < 3; ++c) {
            u64 gsrc = (u64)(uintptr_t)(bpts + (size_t)c * N_ + baseN);
#pragma unroll
            for (int i = 0; i < 4; ++i) {
                unsigned chunk = (unsigned)tid + (unsigned)i * WGS;   // [0,4096)
                u64 ga = gsrc + (u64)chunk * 16u;
                unsigned la = slab_lds + (unsigned)c * (PTS_PER_WG * 4u) + chunk * 16u;
                asm volatile("global_load_async_to_lds_b128 %0, %1, off"
                             :: "v"(la), "v"(ga) : "memory");
            }
        }
        asm volatile("s_wait_asynccnt 0x0" ::: "memory");
    }
    __syncthreads();

    // ---- fan out to registers (stride-1024 LDS reads: conflict free) ----
    float px[PTS_PER_TH], py[PTS_PER_TH], pz[PTS_PER_TH], dst[PTS_PER_TH];
    const float INF = __builtin_huge_valf();
#pragma unroll
    for (int k = 0; k < PTS_PER_TH; ++k) {
        int li = tid + k * WGS;
        px[k] = slab[li];
        py[k] = slab[PTS_PER_WG + li];
        pz[k] = slab[2 * PTS_PER_WG + li];
        dst[k] = INF;
    }

    // initial centroid = point 0 of the batch (reference convention)
    if (tid == 0) {
        scent[0] = bpts[0];
        scent[1] = bpts[N_];
        scent[2] = bpts[2 * N_];
    }
    __syncthreads();

    u64* bbest   = best + (size_t)b * P_;
    u32* barrive = arrive + b * 16;          // 64B-strided counters

    for (int t = 0; t < P_ - 1; ++t) {
        const float cx = scent[0], cy = scent[1], cz = scent[2];
        u64 key = 0ull;
#pragma unroll
        for (int k = 0; k < PTS_PER_TH; ++k) {
            float dx = px[k] - cx;
            float dy = py[k] - cy;
            float dz = pz[k] - cz;
            float d  = __fmaf_rn(dx, dx, __fmaf_rn(dy, dy, dz * dz));
            float nd = fminf(dst[k], d);
            dst[k]   = nd;
            // pack: larger dist wins; on ties, smaller global index wins (~gi larger)
            u32 gi = (u32)(baseN + tid + k * WGS);
            u64 cand = ((u64)__float_as_uint(nd) << 32) | (u64)(u32)(~gi);
            key = kmax_u64(key, cand);
        }
        // wave32 reduction
#pragma unroll
        for (int o = 16; o > 0; o >>= 1)
            key = kmax_u64(key, shfl_xor_u64(key, o));
        if ((tid & 31) == 0) wred[tid >> 5] = key;
        __syncthreads();
        if (tid < 32) {
            u64 k2 = wred[tid];
#pragma unroll
            for (int o = 16; o > 0; o >>= 1)
                k2 = kmax_u64(k2, shfl_xor_u64(k2, o));
            if (tid == 0) {
                atomicMax(&bbest[t + 1], k2);
                __threadfence();
                atomicAdd(barrive, 1u);
                const u32 need = (u32)(WGPB * (t + 1));
                while (__hip_atomic_load(barrive, __ATOMIC_ACQUIRE,
                                         __HIP_MEMORY_SCOPE_AGENT) < need) {
                    __builtin_amdgcn_s_sleep(2);
                }
                u64 kb = __hip_atomic_load(&bbest[t + 1], __ATOMIC_ACQUIRE,
                                           __HIP_MEMORY_SCOPE_AGENT);
                u32 ci = ~(u32)kb;
                scent[0] = bpts[ci];
                scent[1] = bpts[N_ + ci];
                scent[2] = bpts[2 * N_ + ci];
            }
        }
        __syncthreads();
    }
}

// ---------------------------------------------------------------------------
// Gather: out[b, c, t] = points[b, c, idx[b, t]], idx decoded from packed keys
// ---------------------------------------------------------------------------
__global__ void fps_gather(const float* __restrict__ pts,
                           const u64* __restrict__ best,
                           float* __restrict__ out) {
    int j = blockIdx.x * blockDim.x + threadIdx.x;
    if (j >= B_ * P_) return;
    int b = j / P_;
    int t = j - b * P_;
    u32 ci = (t == 0) ? 0u : (u32)(~(u32)best[(size_t)b * P_ + t]);
    const float* bp = pts + (size_t)b * 3 * N_;
    float* bo = out + (size_t)b * 3 * P_;
    bo[t]            = bp[ci];
    bo[P_ + t]       = bp[N_ + ci];
    bo[2 * P_ + t]   = bp[2 * N_ + ci];
}

extern "C" void kernel_launch(void* const* d_in, const int* in_sizes, int n_in,
                              void* d_out, int out_size, void* d_ws, size_t ws_size,
                              hipStream_t stream) {
    (void)in_sizes; (void)n_in; (void)out_size; (void)ws_size;
    const float* pts = (const float*)d_in[0];
    float* out = (float*)d_out;

    // workspace layout: [B*P u64 best keys][B counters, 64B strided]
    u64* best   = (u64*)d_ws;
    u32* arrive = (u32*)((char*)d_ws + (size_t)B_ * P_ * sizeof(u64));

    fps_init<<<(B_ * P_ + 255) / 256, 256, 0, stream>>>(best, arrive);
    fps_kernel<<<B_ * WGPB, WGS, 0, stream>>>(pts, best, arrive);
    fps_gather<<<(B_ * P_ + 255) / 256, 256, 0, stream>>>(pts, best, out);
}